// MultiQueryAttentionSequentialBatched_19756849561624
// MI455X (gfx1250) — compile-verified
//
#include <hip/hip_runtime.h>
#include <hip/hip_bf16.h>

// MQA sequential batched: B=64, H=32, D=4096, K=V=128, M=4095 (Mtot=4096).
// HBM-bound. Strategy:
//  - all GEMMs via v_wmma_f32_16x16x4_f32 (fp32 in / fp32 acc)
//  - concat via b128 copies
//  - launch ORDER exploits the 192MB L2: Kc (134MB) is written immediately
//    before the logits pass consumes it; Vc (134MB) is written immediately
//    before the o-GEMM consumes it. Each fits L2 alone -> prev_K/prev_V are
//    read from HBM exactly once and Kc/Vc are re-read from L2, not HBM.
//    ~670MB HBM traffic -> ~29us roofline at 23.3 TB/s.

#define Bsz  64
#define Hn   32
#define Dd   4096
#define Kdim 128
#define Vdim 128
#define Mold 4095
#define Mtot 4096

typedef float v2f __attribute__((ext_vector_type(2)));
typedef float v8f __attribute__((ext_vector_type(8)));

__device__ __forceinline__ v8f wmma_f32x4(v2f a, v2f b, v8f c) {
    // 8 args: (neg_a, A, neg_b, B, c_mod, C, reuse_a, reuse_b)
    return __builtin_amdgcn_wmma_f32_16x16x4_f32(false, a, false, b,
                                                 (short)0, c, false, false);
}

// ---------------------------------------------------------------- k/v proj
__global__ void __launch_bounds__(256)
proj_kv_kernel(const float* __restrict__ x, const float* __restrict__ Wk,
               const float* __restrict__ Wv, float* __restrict__ knew,
               float* __restrict__ vnew) {
    int b = blockIdx.x;
    int t = threadIdx.x;
    const float* xr = x + (long)b * Dd;
    if (t < Kdim) {
        float acc = 0.f;
        #pragma unroll 4
        for (int d = 0; d < Dd; ++d) acc += xr[d] * Wk[(long)d * Kdim + t];
        knew[(long)b * Kdim + t] = acc;
    } else {
        int c = t - Kdim;
        float acc = 0.f;
        #pragma unroll 4
        for (int d = 0; d < Dd; ++d) acc += xr[d] * Wv[(long)d * Vdim + c];
        vnew[(long)b * Vdim + c] = acc;
    }
}

// ------------------------------------------------------- concat (b128 copy)
// out: [B, Mtot, 128] f32; prev: [B, Mold, 128]; neww: [B, 128]
__global__ void __launch_bounds__(256)
concat_kernel(const float4* __restrict__ prev, const float4* __restrict__ neww,
              float4* __restrict__ out) {
    long idx = (long)blockIdx.x * blockDim.x + threadIdx.x;
    const long total = (long)Bsz * Mtot * (Kdim / 4);
    if (idx >= total) return;
    int  c  = (int)(idx & 31);           // float4 column (128/4 = 32)
    long bm = idx >> 5;
    int  m  = (int)(bm % Mtot);
    int  b  = (int)(bm / Mtot);
    float4 v = (m < Mold) ? prev[((long)b * Mold + m) * 32 + c]
                          : neww[(long)b * 32 + c];
    out[idx] = v;
}

// --------------------------------------------------- q = x @ Wq  (64x4096x4096)
// C[m=batch 64, n=h*128+k 4096]; B[d,n] = Wq[h, d, k] (d-stride = 128)
__global__ void __launch_bounds__(256)
qproj_kernel(const float* __restrict__ x, const float* __restrict__ Wq,
             float* __restrict__ qout) {
    int tile = blockIdx.x * 8 + (threadIdx.x >> 5);
    int lane = threadIdx.x & 31;
    if (tile >= 4 * 256) return;
    int tm = tile & 3;            // batch tile (64/16)
    int tn = tile >> 2;           // col tile (4096/16)
    int l16  = lane & 15;
    int row  = tm * 16 + l16;                 // A row this lane feeds
    int nc   = tn * 16 + l16;                 // B col this lane feeds
    int h    = nc >> 7;
    int kk   = nc & 127;
    int klo  = (lane < 16) ? 0 : 2;
    const float2* Arow = (const float2*)(x + (long)row * Dd);
    const float*  Bcol = Wq + (long)h * Dd * Kdim + kk;   // index by d*128
    v8f c = {};
    #pragma unroll 4
    for (int k0 = 0; k0 < Dd; k0 += 4) {
        float2 af = Arow[(k0 + klo) >> 1];
        v2f a; a.x = af.x; a.y = af.y;
        v2f b; b.x = Bcol[(long)(k0 + klo) * Kdim];
               b.y = Bcol[(long)(k0 + klo + 1) * Kdim];
        c = wmma_f32x4(a, b, c);
    }
    int ccol   = tn * 16 + l16;
    int rowoff = tm * 16 + ((lane < 16) ? 0 : 8);
    #pragma unroll
    for (int r = 0; r < 8; ++r)
        qout[(long)(rowoff + r) * 4096 + ccol] = c[r];
}

// ------------------------------------- logits[b,h,m] = sum_k q[b,h,k]*Kc[b,m,k]
// One wave per (b, m-tile); dual accumulators cover all 32 heads so the Kc
// B-fragment is loaded once and feeds two WMMAs per k-step.
__global__ void __launch_bounds__(256)
logits_kernel(const float* __restrict__ q, const float* __restrict__ Kc,
              float* __restrict__ logits) {
    int tile = blockIdx.x * 8 + (threadIdx.x >> 5);
    int lane = threadIdx.x & 31;
    const int total = Bsz * (Mtot / 16);
    if (tile >= total) return;
    int b  = tile / (Mtot / 16);
    int tm = tile % (Mtot / 16);
    int l16  = lane & 15;
    int mcol = tm * 16 + l16;
    int klo  = (lane < 16) ? 0 : 2;
    const float2* Aq0 = (const float2*)(q  + ((long)b * Hn + l16)      * Kdim);
    const float2* Aq1 = (const float2*)(q  + ((long)b * Hn + 16 + l16) * Kdim);
    const float2* Bk  = (const float2*)(Kc + ((long)b * Mtot + mcol)   * Kdim);
    v8f c0 = {}, c1 = {};
    #pragma unroll
    for (int k0 = 0; k0 < Kdim; k0 += 4) {
        int e = (k0 + klo) >> 1;
        float2 bf = Bk[e];
        float2 a0 = Aq0[e];
        float2 a1 = Aq1[e];
        v2f bb; bb.x = bf.x; bb.y = bf.y;
        v2f va0; va0.x = a0.x; va0.y = a0.y;
        v2f va1; va1.x = a1.x; va1.y = a1.y;
        c0 = wmma_f32x4(va0, bb, c0);
        c1 = wmma_f32x4(va1, bb, c1);
    }
    int rowoff = (lane < 16) ? 0 : 8;
    #pragma unroll
    for (int r8 = 0; r8 < 8; ++r8) {
        int hh = rowoff + r8;
        long base = ((long)b * Hn + hh) * Mtot + tm * 16 + l16;
        logits[base]              = c0[r8];
        logits[base + 16L * Mtot] = c1[r8];
    }
}

// ----------------------------------------------------- softmax rows of 4096
__global__ void __launch_bounds__(256)
softmax_kernel(float* __restrict__ logits) {
    long row = blockIdx.x;               // b*H + h (2048 rows)
    float* p = logits + row * Mtot;
    __shared__ float red[256];
    int t = threadIdx.x;
    float mx = -INFINITY;
    for (int i = t; i < Mtot; i += 256) mx = fmaxf(mx, p[i]);
    red[t] = mx; __syncthreads();
    for (int s = 128; s > 0; s >>= 1) {
        if (t < s) red[t] = fmaxf(red[t], red[t + s]);
        __syncthreads();
    }
    mx = red[0]; __syncthreads();
    float sum = 0.f;
    for (int i = t; i < Mtot; i += 256) {
        float e = __expf(p[i] - mx);
        p[i] = e;
        sum += e;
    }
    red[t] = sum; __syncthreads();
    for (int s = 128; s > 0; s >>= 1) {
        if (t < s) red[t] += red[t + s];
        __syncthreads();
    }
    float inv = 1.0f / red[0];
    for (int i = t; i < Mtot; i += 256) p[i] *= inv;
}

// --------------------------------------- o[b,h,v] = sum_m w[b,h,m]*Vc[b,m,v]
__global__ void __launch_bounds__(256)
ogemm_kernel(const float* __restrict__ w, const float* __restrict__ Vc,
             float* __restrict__ o) {
    int tile = blockIdx.x * 8 + (threadIdx.x >> 5);
    int lane = threadIdx.x & 31;
    const int total = Bsz * 2 * (Vdim / 16);
    if (tile >= total) return;
    int b  = tile / 16;
    int r  = tile % 16;
    int th = r >> 3;              // head tile
    int tv = r & 7;               // v tile
    int l16  = lane & 15;
    int hrow = th * 16 + l16;
    int vcol = tv * 16 + l16;
    int klo  = (lane < 16) ? 0 : 2;
    const float2* Aw = (const float2*)(w + ((long)b * Hn + hrow) * Mtot);
    const float*  Bv = Vc + (long)b * Mtot * Vdim + vcol;   // index by m*128
    v8f c = {};
    #pragma unroll 4
    for (int m0 = 0; m0 < Mtot; m0 += 4) {
        float2 af = Aw[(m0 + klo) >> 1];
        v2f a; a.x = af.x; a.y = af.y;
        v2f b2; b2.x = Bv[(long)(m0 + klo) * Vdim];
                b2.y = Bv[(long)(m0 + klo + 1) * Vdim];
        c = wmma_f32x4(a, b2, c);
    }
    int rowoff = (lane < 16) ? 0 : 8;
    #pragma unroll
    for (int r8 = 0; r8 < 8; ++r8) {
        int hh = th * 16 + rowoff + r8;
        o[((long)b * Hn + hh) * Vdim + tv * 16 + l16] = c[r8];
    }
}

// ------------------------ y = o_flat[64,4096] @ Wo_flat[4096,4096] (row-major)
__global__ void __launch_bounds__(256)
ygemm_kernel(const float* __restrict__ o, const float* __restrict__ Wo,
             float* __restrict__ y) {
    int tile = blockIdx.x * 8 + (threadIdx.x >> 5);
    int lane = threadIdx.x & 31;
    if (tile >= 4 * 256) return;
    int tm = tile & 3;            // batch tile
    int tn = tile >> 2;           // d tile
    int l16 = lane & 15;
    int row = tm * 16 + l16;
    int nc  = tn * 16 + l16;
    int klo = (lane < 16) ? 0 : 2;
    const float2* Arow = (const float2*)(o + (long)row * 4096);
    const float*  Bcol = Wo + nc;                 // index by k*4096
    v8f c = {};
    #pragma unroll 4
    for (int k0 = 0; k0 < 4096; k0 += 4) {
        float2 af = Arow[(k0 + klo) >> 1];
        v2f a; a.x = af.x; a.y = af.y;
        v2f b; b.x = Bcol[(long)(k0 + klo) * 4096];
               b.y = Bcol[(long)(k0 + klo + 1) * 4096];
        c = wmma_f32x4(a, b, c);
    }
    int ccol   = tn * 16 + l16;
    int rowoff = tm * 16 + ((lane < 16) ? 0 : 8);
    #pragma unroll
    for (int r = 0; r < 8; ++r)
        y[(long)(rowoff + r) * Dd + ccol] = c[r];
}

extern "C" void kernel_launch(void* const* d_in, const int* in_sizes, int n_in,
                              void* d_out, int out_size, void* d_ws, size_t ws_size,
                              hipStream_t stream) {
    const float* x      = (const float*)d_in[0];
    const float* prev_K = (const float*)d_in[1];
    const float* prev_V = (const float*)d_in[2];
    const float* Wq     = (const float*)d_in[3];
    const float* Wk     = (const float*)d_in[4];
    const float* Wv     = (const float*)d_in[5];
    const float* Wo     = (const float*)d_in[6];

    float* y  = (float*)d_out;                      // [64, 4096]
    float* Kc = y  + (long)Bsz * Dd;                // [64, 4096, 128]
    float* Vc = Kc + (long)Bsz * Mtot * Kdim;       // [64, 4096, 128]

    float* ws     = (float*)d_ws;
    float* q      = ws;                             // 64*32*128   = 262144
    float* knew   = q      + (long)Bsz * Hn * Kdim; // 64*128      = 8192
    float* vnew   = knew   + (long)Bsz * Kdim;      // 64*128      = 8192
    float* logits = vnew   + (long)Bsz * Vdim;      // 64*32*4096  = 8388608
    float* o      = logits + (long)Bsz * Hn * Mtot; // 64*32*128   = 262144

    long cpy_total = (long)Bsz * Mtot * (Kdim / 4);
    int  cpy_blocks = (int)((cpy_total + 255) / 256);

    // 1. k_new / v_new projections (tiny)
    proj_kv_kernel<<<Bsz, 256, 0, stream>>>(x, Wk, Wv, knew, vnew);

    // 2. q projection FIRST: streams x+Wq (67MB) before Kc occupies L2
    qproj_kernel<<<(4 * 256) / 8, 256, 0, stream>>>(x, Wq, q);

    // 3. concat K -> Kc (134MB write; stays L2-resident, < 192MB L2)
    concat_kernel<<<cpy_blocks, 256, 0, stream>>>((const float4*)prev_K,
                                                  (const float4*)knew,
                                                  (float4*)Kc);

    // 4. logits: reads Kc from L2; 64*256 = 16384 dual-head waves
    logits_kernel<<<(Bsz * (Mtot / 16)) / 8, 256, 0, stream>>>(q, Kc, logits);

    // 5. softmax: 2048 rows, logits (33MB) L2-hot
    softmax_kernel<<<Bsz * Hn, 256, 0, stream>>>(logits);

    // 6. concat V -> Vc (Kc dead; Vc takes over L2)
    concat_kernel<<<cpy_blocks, 256, 0, stream>>>((const float4*)prev_V,
                                                  (const float4*)vnew,
                                                  (float4*)Vc);

    // 7. o = weights @ Vc: Vc served from L2; 1024 waves
    ogemm_kernel<<<(Bsz * 2 * (Vdim / 16)) / 8, 256, 0, stream>>>(logits, Vc, o);

    // 8. y = o @ Wo: streams Wo last
    ygemm_kernel<<<(4 * 256) / 8, 256, 0, stream>>>(o, Wo, y);
}